// attention_78151224918786
// MI455X (gfx1250) — compile-verified
//
#include <hip/hip_runtime.h>
#include <math.h>

// Problem sizes (fixed by the reference)
#define N_ROWS  200000
#define IN_DIM  512
#define OUT_DIM 256

// Launch geometry for the streaming pass
#define POOL_BLOCKS 256
#define POOL_TPB    256              // 8 waves of 32
#define WAVES_PER_BLOCK (POOL_TPB / 32)
#define PARTIAL_STRIDE (OUT_DIM + 2) // [acc0..acc255, M, Z]

typedef float v2f __attribute__((ext_vector_type(2)));
typedef float v8f __attribute__((ext_vector_type(8)));

// ---------------------------------------------------------------------------
// Kernel 1: q = W @ att + bias via V_WMMA_F32_16X16X4_F32 (broadcast-B matvec)
// 1 block x 512 threads = 16 waves; wave w computes q[16w .. 16w+15].
// A tile (16x4 f32): lanes 0-15 hold K={0,1} in VGPR{0,1}; lanes 16-31 K={2,3}.
// B tile (4x16 f32): every column n holds the same att[] chunk, so every
// column of D accumulates the matvec result.
// ---------------------------------------------------------------------------
__global__ __launch_bounds__(512) void attnpool_proj_q_wmma(
    const float* __restrict__ att,   // [512]
    const float* __restrict__ W,     // [256][512] row-major
    const float* __restrict__ bias,  // [256]
    float* __restrict__ q)           // [256] out
{
  const int tid  = threadIdx.x;
  const int lane = tid & 31;
  const int wave = tid >> 5;           // 0..15
  const int m    = lane & 15;          // row within tile
  const int khi  = (lane >> 4) << 1;   // 0 for lanes 0-15, 2 for lanes 16-31
  const int row  = wave * 16 + m;
  const float* wrow = W + (size_t)row * IN_DIM;

  v8f acc = {0.f, 0.f, 0.f, 0.f, 0.f, 0.f, 0.f, 0.f};

  for (int k0 = 0; k0 < IN_DIM; k0 += 4) {
    v2f a, bb;
    a.x  = wrow[k0 + khi];
    a.y  = wrow[k0 + khi + 1];
    bb.x = att[k0 + khi];
    bb.y = att[k0 + khi + 1];
    // D = A x B + C  (8 args: neg_a, A, neg_b, B, c_mod, C, reuse_a, reuse_b)
    acc = __builtin_amdgcn_wmma_f32_16x16x4_f32(
        false, a, false, bb, (short)0, acc, false, false);
  }

  // D layout: VGPR r -> M=r (lanes 0-15), M=8+r (lanes 16-31); all N equal.
  if (lane == 0) {
#pragma unroll
    for (int r = 0; r < 8; ++r) q[wave * 16 + r] = acc[r] + bias[wave * 16 + r];
  } else if (lane == 16) {
#pragma unroll
    for (int r = 0; r < 8; ++r) q[wave * 16 + 8 + r] = acc[r] + bias[wave * 16 + 8 + r];
  }
}

// ---------------------------------------------------------------------------
// Kernel 2: single-pass online-softmax pooling.
// Each wave grid-strides over rows. Per row (256 f32 = 1 KB):
//   lane l loads cols [4l..4l+3] and [128+4l..128+4l+3] (two b128 loads),
//   dot with preloaded q, wave-reduce -> score p (wave-uniform),
//   readfirstlane(p) -> scalar branch:
//     common path (max unchanged): e = e^(p-m); Z += e; acc += e*x
//     rare path  (new max):        c = e^(m-p); Z = Z*c + 1; acc = acc*c + x
// The row is touched exactly once (score + pooling fused) -> ~8.8 us floor.
// Wave partials combine in LDS to one block partial [acc(256), M, Z].
// ---------------------------------------------------------------------------
__global__ __launch_bounds__(POOL_TPB) void attnpool_online_pool(
    const float* __restrict__ x,        // [N][256]
    const float* __restrict__ q,        // [256]
    float* __restrict__ partials)       // [POOL_BLOCKS][PARTIAL_STRIDE]
{
  const int tid    = threadIdx.x;
  const int lane   = tid & 31;
  const int w      = tid >> 5;
  const int gwave  = blockIdx.x * WAVES_PER_BLOCK + w;
  const int nwaves = gridDim.x * WAVES_PER_BLOCK;

  const float4* q4 = (const float4*)q;
  const float4 q0 = q4[lane];        // cols 4l..4l+3
  const float4 q1 = q4[lane + 32];   // cols 128+4l..128+4l+3

  float  m = -INFINITY;
  float  Z = 0.f;
  float4 a0 = {0.f, 0.f, 0.f, 0.f};
  float4 a1 = {0.f, 0.f, 0.f, 0.f};

  for (int row = gwave; row < N_ROWS; row += nwaves) {
    const float4* xr = (const float4*)(x + (size_t)row * OUT_DIM);
    // Prefetch next grid-stride row (global_prefetch_b8; speculative, OOB-safe)
    __builtin_prefetch((const char*)x + ((size_t)(row + nwaves)) * (OUT_DIM * 4)
                           + lane * 32, 0, 0);
    const float4 x0 = xr[lane];
    const float4 x1 = xr[lane + 32];

    float p = x0.x * q0.x + x0.y * q0.y + x0.z * q0.z + x0.w * q0.w
            + x1.x * q1.x + x1.y * q1.y + x1.z * q1.z + x1.w * q1.w;
    // wave32 cross-lane sum -> same value in all lanes
#pragma unroll
    for (int off = 16; off > 0; off >>= 1) p += __shfl_xor(p, off, 32);
    // Make uniformity explicit so the max-update test is a scalar branch.
    p = __builtin_amdgcn_readfirstlane(p);

    if (p > m) {
      // Rare path (~ln(rows) times per wave). First row: c = exp(-inf) = 0.
      const float c = __expf(m - p);
      Z = Z * c + 1.0f;
      a0.x = a0.x * c + x0.x;  a0.y = a0.y * c + x0.y;
      a0.z = a0.z * c + x0.z;  a0.w = a0.w * c + x0.w;
      a1.x = a1.x * c + x1.x;  a1.y = a1.y * c + x1.y;
      a1.z = a1.z * c + x1.z;  a1.w = a1.w * c + x1.w;
      m = p;
    } else {
      // Common path: 1 transcendental + 8 FMAs per 1 KB row.
      const float e = __expf(p - m);
      Z += e;
      a0.x += e * x0.x;  a0.y += e * x0.y;
      a0.z += e * x0.z;  a0.w += e * x0.w;
      a1.x += e * x1.x;  a1.y += e * x1.y;
      a1.z += e * x1.z;  a1.w += e * x1.w;
    }
  }

  // ---- block combine in LDS ----
  __shared__ float sm[WAVES_PER_BLOCK];
  __shared__ float sz[WAVES_PER_BLOCK];
  __shared__ float sacc[WAVES_PER_BLOCK][OUT_DIM];

  sacc[w][4 * lane + 0]       = a0.x;
  sacc[w][4 * lane + 1]       = a0.y;
  sacc[w][4 * lane + 2]       = a0.z;
  sacc[w][4 * lane + 3]       = a0.w;
  sacc[w][128 + 4 * lane + 0] = a1.x;
  sacc[w][128 + 4 * lane + 1] = a1.y;
  sacc[w][128 + 4 * lane + 2] = a1.z;
  sacc[w][128 + 4 * lane + 3] = a1.w;
  if (lane == 0) { sm[w] = m; sz[w] = Z; }
  __syncthreads();

  // thread t owns output column t
  float M = sm[0];
#pragma unroll
  for (int i = 1; i < WAVES_PER_BLOCK; ++i) M = fmaxf(M, sm[i]);

  float Zb = 0.f, ab = 0.f;
#pragma unroll
  for (int i = 0; i < WAVES_PER_BLOCK; ++i) {
    const float s = __expf(sm[i] - M);
    Zb += sz[i] * s;
    ab += sacc[i][tid] * s;
  }

  float* bp = partials + (size_t)blockIdx.x * PARTIAL_STRIDE;
  bp[tid] = ab;
  if (tid == 0) { bp[OUT_DIM] = M; bp[OUT_DIM + 1] = Zb; }
}

// ---------------------------------------------------------------------------
// Kernel 3: final reduction across block partials (tiny: ~266 KB in L2).
// ---------------------------------------------------------------------------
__global__ __launch_bounds__(OUT_DIM) void attnpool_finalize(
    const float* __restrict__ partials,  // [nblocks][PARTIAL_STRIDE]
    float* __restrict__ out,             // [256]
    int nblocks)
{
  const int t = threadIdx.x;

  float M = -INFINITY;
  for (int b = 0; b < nblocks; ++b)
    M = fmaxf(M, partials[(size_t)b * PARTIAL_STRIDE + OUT_DIM]);

  float Z = 0.f, acc = 0.f;
  for (int b = 0; b < nblocks; ++b) {
    const float* bp = partials + (size_t)b * PARTIAL_STRIDE;
    const float s = __expf(bp[OUT_DIM] - M);
    Z   += bp[OUT_DIM + 1] * s;
    acc += bp[t] * s;
  }
  out[t] = acc / Z;
}

// ---------------------------------------------------------------------------
// Host-side launch. Input order: att_val[512], input[N*256], W[256*512], b[256]
// Workspace layout: q at offset 0 (1 KB), block partials at offset 1024.
// ---------------------------------------------------------------------------
extern "C" void kernel_launch(void* const* d_in, const int* in_sizes, int n_in,
                              void* d_out, int out_size, void* d_ws, size_t ws_size,
                              hipStream_t stream) {
  const float* att   = (const float*)d_in[0];
  const float* input = (const float*)d_in[1];
  const float* W     = (const float*)d_in[2];
  const float* bias  = (const float*)d_in[3];
  float* out = (float*)d_out;

  float* q        = (float*)d_ws;                       // 256 floats
  float* partials = (float*)((char*)d_ws + 1024);       // POOL_BLOCKS * 258 floats

  attnpool_proj_q_wmma<<<1, 512, 0, stream>>>(att, W, bias, q);
  attnpool_online_pool<<<POOL_BLOCKS, POOL_TPB, 0, stream>>>(input, q, partials);
  attnpool_finalize<<<1, OUT_DIM, 0, stream>>>(partials, out, POOL_BLOCKS);
}